// GaussKernel_54992761258600
// MI455X (gfx1250) — compile-verified
//
#include <hip/hip_runtime.h>
#include <math.h>

// Problem constants (from reference setup_inputs)
#define NN 8192   // rows of c_in
#define KK 8      // list length
#define MM 2048   // centers
#define CC 10     // classes
#define KCP 84    // K*C (=80) + 1 (psq+K column) padded to multiple of 4

typedef __attribute__((ext_vector_type(2))) float v2f;
typedef __attribute__((ext_vector_type(8))) float v8f;

// ---------------------------------------------------------------------------
// prep_A: softmax over c_p[k][m][:], build A row (f32, 84 wide):
//   A[m][k*10+c] = -2 * softmax(c_p[k,m,:])[c]
//   A[m][80]     = sum_k sum_c p^2 + K      (so GEMM result == metric)
//   A[m][81..83] = 0
// also zero-init the per-m normalizer S[m].
// ---------------------------------------------------------------------------
__global__ void prep_A(const float* __restrict__ c_p,
                       float* __restrict__ A,
                       float* __restrict__ S) {
  int m = blockIdx.x * blockDim.x + threadIdx.x;
  if (m >= MM) return;
  float* arow = A + (size_t)m * KCP;
  float psq = 0.0f;
  for (int k = 0; k < KK; ++k) {
    const float* row = c_p + ((size_t)k * MM + (size_t)m) * CC;
    float v[CC];
    float mx = -INFINITY;
#pragma unroll
    for (int c = 0; c < CC; ++c) { v[c] = row[c]; mx = fmaxf(mx, v[c]); }
    float sum = 0.0f;
#pragma unroll
    for (int c = 0; c < CC; ++c) { v[c] = expf(v[c] - mx); sum += v[c]; }
    float inv = 1.0f / sum;
#pragma unroll
    for (int c = 0; c < CC; ++c) {
      float p = v[c] * inv;
      psq += p * p;
      arow[k * CC + c] = -2.0f * p;
    }
  }
  arow[80] = psq + (float)KK;
  arow[81] = 0.0f;
  arow[82] = 0.0f;
  arow[83] = 0.0f;
  S[m] = 0.0f;
}

// ---------------------------------------------------------------------------
// prep_B: one-hot row of c_in (f32, 84 wide): B[n][k*10+c_in[n,k]] = 1,
// B[n][80] = 1 (multiplies the psq+K column), rest 0.
// ---------------------------------------------------------------------------
__global__ void prep_B(const int* __restrict__ c_in,
                       float* __restrict__ B) {
  int n = blockIdx.x * blockDim.x + threadIdx.x;
  if (n >= NN) return;
  float* brow = B + (size_t)n * KCP;
#pragma unroll
  for (int i = 0; i < KCP; ++i) brow[i] = 0.0f;
#pragma unroll
  for (int k = 0; k < KK; ++k) {
    int c = c_in[(size_t)n * KK + k];
    if (c >= 0 && c < CC) brow[k * CC + c] = 1.0f;
  }
  brow[80] = 1.0f;
}

// ---------------------------------------------------------------------------
// One 16(m) x 16(n) metric tile via 21 chained V_WMMA_F32_16X16X4_F32.
// A fragment (16x4): lanes 0-15 -> M=lane, VGPR0/1 = K0/K1; lanes 16-31 -> K2/K3.
// B fragment (4x16): mirrored (VGPR index = inner K, +2 for upper lane half).
// Both load as aligned float2 from row-major 84-wide rows.
// ---------------------------------------------------------------------------
__device__ __forceinline__ v8f tile_metric(const float* __restrict__ A,
                                           const float* __restrict__ B,
                                           int m0, int n0, int lane) {
  const int half = lane >> 4;
  const int l15  = lane & 15;
  const float* arow = A + (size_t)(m0 + l15) * KCP + 2 * half;
  const float* brow = B + (size_t)(n0 + l15) * KCP + 2 * half;
  v8f acc = {0.f, 0.f, 0.f, 0.f, 0.f, 0.f, 0.f, 0.f};
#pragma unroll
  for (int kb = 0; kb < KCP; kb += 4) {
    v2f a = *(const v2f*)(arow + kb);
    v2f b = *(const v2f*)(brow + kb);
    acc = __builtin_amdgcn_wmma_f32_16x16x4_f32(false, a, false, b,
                                                (short)0, acc, false, false);
  }
  return acc;  // acc[r] = metric[m0 + r + 8*half][n0 + l15]
}

// ---------------------------------------------------------------------------
// Pass 1: S[m] += sum_n exp(-metric/bw).  Block = 8 waves covering 32m x 64n.
// ---------------------------------------------------------------------------
__global__ void pass_sum(const float* __restrict__ A,
                         const float* __restrict__ B,
                         const float* __restrict__ bwp,
                         float* __restrict__ S) {
  const int lane = threadIdx.x & 31;
  const int wave = threadIdx.x >> 5;
  const int m0 = blockIdx.y * 32 + (wave >> 2) * 16;
  const int n0 = blockIdx.x * 64 + (wave & 3) * 16;
  const float bw = fminf(fmaxf(bwp[0], 0.1f), 10.0f);
  const float ninv = -1.0f / bw;

  v8f acc = tile_metric(A, B, m0, n0, lane);

  const int half = lane >> 4;
  const int l15  = lane & 15;
#pragma unroll
  for (int r = 0; r < 8; ++r) {
    float w = expf(acc[r] * ninv);
    w += __shfl_xor(w, 1, 32);
    w += __shfl_xor(w, 2, 32);
    w += __shfl_xor(w, 4, 32);
    w += __shfl_xor(w, 8, 32);   // sum over the 16 lanes sharing this half
    if (l15 == 0) atomicAdd(&S[m0 + r + 8 * half], w);
  }
}

// ---------------------------------------------------------------------------
// Pass 2: recompute metric, write out[m][n] = exp(-metric/bw) / S[m]
// (0 where S[m] < 1e-13).  Output layout [M, N, 1] row-major.
// ---------------------------------------------------------------------------
__global__ void pass_out(const float* __restrict__ A,
                         const float* __restrict__ B,
                         const float* __restrict__ bwp,
                         const float* __restrict__ S,
                         float* __restrict__ out) {
  const int lane = threadIdx.x & 31;
  const int wave = threadIdx.x >> 5;
  const int m0 = blockIdx.y * 32 + (wave >> 2) * 16;
  const int n0 = blockIdx.x * 64 + (wave & 3) * 16;
  const float bw = fminf(fmaxf(bwp[0], 0.1f), 10.0f);
  const float ninv = -1.0f / bw;

  v8f acc = tile_metric(A, B, m0, n0, lane);

  const int half = lane >> 4;
  const int l15  = lane & 15;
#pragma unroll
  for (int r = 0; r < 8; ++r) {
    const int m = m0 + r + 8 * half;
    const float s = S[m];
    const float scale = (s < 1e-13f) ? 0.0f : (1.0f / s);
    const float w = expf(acc[r] * ninv) * scale;
    out[(size_t)m * NN + (size_t)(n0 + l15)] = w;
  }
}

// ---------------------------------------------------------------------------
extern "C" void kernel_launch(void* const* d_in, const int* in_sizes, int n_in,
                              void* d_out, int out_size, void* d_ws, size_t ws_size,
                              hipStream_t stream) {
  (void)in_sizes; (void)n_in; (void)out_size; (void)ws_size;
  const int*   c_in = (const int*)d_in[0];    // [N, K] labels
  const float* c_p  = (const float*)d_in[1];  // [K, M, C] logits
  const float* bwp  = (const float*)d_in[2];  // [1] bandwidth
  float* out = (float*)d_out;                 // [M, N, 1]

  // Workspace: A (M*84 f32) | B (N*84 f32) | S (M f32)  ~= 3.5 MB
  float* A = (float*)d_ws;
  float* B = A + (size_t)MM * KCP;
  float* S = B + (size_t)NN * KCP;

  prep_A<<<(MM + 255) / 256, 256, 0, stream>>>(c_p, A, S);
  prep_B<<<(NN + 255) / 256, 256, 0, stream>>>(c_in, B);

  dim3 grid(NN / 64, MM / 32);  // 128 x 64 blocks, 256 threads (8 waves) each
  pass_sum<<<grid, 256, 0, stream>>>(A, B, bwp, S);
  pass_out<<<grid, 256, 0, stream>>>(A, B, bwp, S, out);
}